// QKProjectionLayer_54176717471932
// MI455X (gfx1250) — compile-verified
//
#include <hip/hip_runtime.h>
#include <math.h>

#define B_ 4
#define L_ 1024
#define D_ 256
#define LDSW 260   // padded LDS row stride (floats): conflict-free, float4/b128-aligned
#define SSTR 36    // padded stride for 32x32 score tile

typedef __attribute__((ext_vector_type(2))) float v2f;
typedef __attribute__((ext_vector_type(8))) float v8f;

// -------- CDNA5 async global->LDS path (ASYNCcnt), with portable fallback ----
#if defined(__gfx1250__) && __has_builtin(__builtin_amdgcn_global_load_async_to_lds_b128) && \
    __has_builtin(__builtin_amdgcn_s_wait_asynccnt)
#define HAVE_ASYNC_LDS 1
#else
#define HAVE_ASYNC_LDS 0
#endif

#if HAVE_ASYNC_LDS
typedef int v4i_vs __attribute__((vector_size(16)));
typedef __attribute__((address_space(1))) v4i_vs* g_v4i_ptr;   // global (AS1)
typedef __attribute__((address_space(3))) v4i_vs* l_v4i_ptr;   // LDS (AS3)
#endif

// Copy a 32 x 256 float tile (contiguous global rows) into LDS with row stride LDSW.
__device__ __forceinline__ void copy_tile(const float* __restrict__ src, float* dst, int tid) {
#if HAVE_ASYNC_LDS
  for (int i = tid; i < 32 * 64; i += 256) {
    int r = i >> 6, c4 = (i & 63) * 4;
    __builtin_amdgcn_global_load_async_to_lds_b128(
        (g_v4i_ptr)(src + (size_t)r * D_ + c4),
        (l_v4i_ptr)(dst + r * LDSW + c4),
        0, 0);
  }
#else
  for (int i = tid; i < 32 * 64; i += 256) {
    int r = i >> 6, c4 = (i & 63) * 4;
    *(float4*)(dst + r * LDSW + c4) = *(const float4*)(src + (size_t)r * D_ + c4);
  }
#endif
}

__device__ __forceinline__ void wait_async_all() {
#if HAVE_ASYNC_LDS
  __builtin_amdgcn_s_wait_asynccnt(0);
#endif
}

__device__ __forceinline__ v8f vzero8() {
  v8f z;
  for (int i = 0; i < 8; ++i) z[i] = 0.f;
  return z;
}

// A-fragment (and B-fragment for X @ Y^T) of V_WMMA_F32_16X16X4_F32:
// lanes 0-15 hold row m=l with k = kk,kk+1 ; lanes 16-31 hold k = kk+2,kk+3.
__device__ __forceinline__ v2f frag_rm(const float* p, int ld, int l, int hi, int kk) {
  const float* s = p + l * ld + kk + 2 * hi;
  v2f r; r.x = s[0]; r.y = s[1]; return r;
}
// B-fragment for X @ Y (k-major source): element (k, n=l) at p[k*ld + l]
__device__ __forceinline__ v2f frag_cm(const float* p, int ld, int l, int hi, int kk) {
  v2f r;
  r.x = p[(kk + 2 * hi) * ld + l];
  r.y = p[(kk + 2 * hi + 1) * ld + l];
  return r;
}

#define WMMAF32(a, b, c) \
  __builtin_amdgcn_wmma_f32_16x16x4_f32(false, (a), false, (b), (short)0, (c), false, false)

// ---------------------------------------------------------------------------
// Pass 1: per (batch, 32-row block):
//   acc  = Q_blk @ P_prev^T + tril(Q_blk K^T) @ K         (raw numerator)
//   d[t] = 2*k_t^T P_prev k_t + 2*sum_{r<t} G[t,r]^2 + G[t,t]^2
// K chunks are double-buffered in LDS via GLOBAL_LOAD_ASYNC_TO_LDS_B128.
// ---------------------------------------------------------------------------
__global__ void qkproj_pass1(const float* __restrict__ q, const float* __restrict__ k,
                             const float* __restrict__ P, float* __restrict__ qraw,
                             float* __restrict__ dacc)
{
  extern __shared__ float smem[];
  float* Qs  = smem;                 // 32 x LDSW
  float* Ks  = Qs + 32 * LDSW;       // 32 x LDSW
  float* Kc0 = Ks + 32 * LDSW;       // 32 x LDSW  (K chunk buffer 0)
  float* Kc1 = Kc0 + 32 * LDSW;      // 32 x LDSW  (K chunk buffer 1)
  float* Ssh = Kc1 + 32 * LDSW;      // 32 x SSTR masked scores
  float* dsh = Ssh + 32 * SSTR;      // 32 fro-delta accumulators
  float* esh = dsh + 32;             // 32 k^T P k accumulators

  const int b   = blockIdx.y;
  const int t0  = blockIdx.x * 32;
  const int tid = threadIdx.x;
  const int lane = tid & 31, l = lane & 15, hi = lane >> 4;
  const int w = tid >> 5, rh = w >> 2, cg = w & 3;

  const float* Qb = q + ((size_t)b * L_ + t0) * D_;
  const float* Kb = k + (size_t)b * L_ * D_;
  const float* Pb = P + (size_t)b * D_ * D_;

  // async-stage Q block, K block, and the first K chunk; single wait covers all
  copy_tile(Qb, Qs, tid);
  copy_tile(Kb + (size_t)t0 * D_, Ks, tid);
  copy_tile(Kb, Kc0, tid);
  if (tid < 32) { dsh[tid] = 0.f; esh[tid] = 0.f; }
  wait_async_all();
  __syncthreads();

  v8f acc[4];

  // ---- P_prev phase: acc = Q_blk @ P^T ; esh[t] = k_t^T P_prev k_t partials
  for (int nt = 0; nt < 4; ++nt) {
    const int nb = cg * 64 + nt * 16;
    v8f c = vzero8();
    v8f e = vzero8();
    for (int kk = 0; kk < D_; kk += 4) {
      v2f bf = frag_rm(Pb + (size_t)nb * D_, D_, l, hi, kk);  // B[k=j, n=i] = P[i*D+j]
      c = WMMAF32(frag_rm(Qs + 16 * rh * LDSW, LDSW, l, hi, kk), bf, c);
      e = WMMAF32(frag_rm(Ks + 16 * rh * LDSW, LDSW, l, hi, kk), bf, e);
    }
    acc[nt] = c;
    for (int v = 0; v < 8; ++v) {
      int M = v + 8 * hi;
      atomicAdd(&esh[16 * rh + M], e[v] * Ks[(16 * rh + M) * LDSW + nb + l]);
    }
  }

  // ---- causal chunk loop over K rows [0, t0+32), double-buffered
  float* Kbuf[2] = { Kc0, Kc1 };
  int buf = 0;
  for (int sc = 0; sc <= t0; sc += 32) {
    if (sc) {                       // wait for the chunk prefetched last iteration
      wait_async_all();
      __syncthreads();
    }
    if (sc + 32 <= t0)              // prefetch next chunk into the other buffer
      copy_tile(Kb + (size_t)(sc + 32) * D_, Kbuf[buf ^ 1], tid);
    const float* Kcur = Kbuf[buf];

    // phase A: waves 0-3 -> S = Q_blk Kc^T ; waves 4-7 -> G = K_blk Kc^T
    {
      const int mi = (w & 3) >> 1, ni = w & 1;
      const bool isG = (w >= 4);
      const float* Ar = (isG ? Ks : Qs) + 16 * mi * LDSW;
      const float* Br = Kcur + 16 * ni * LDSW;
      v8f cc = vzero8();
      for (int kk = 0; kk < D_; kk += 4)
        cc = WMMAF32(frag_rm(Ar, LDSW, l, hi, kk), frag_rm(Br, LDSW, l, hi, kk), cc);
      for (int v = 0; v < 8; ++v) {
        const int M = v + 8 * hi;
        const int t = t0 + 16 * mi + M;
        const int s = sc + 16 * ni + l;
        if (!isG) {
          Ssh[(16 * mi + M) * SSTR + 16 * ni + l] = (s <= t) ? cc[v] : 0.f;
        } else {
          float wt = (s < t) ? 2.f : ((s == t) ? 1.f : 0.f);
          if (wt != 0.f) atomicAdd(&dsh[16 * mi + M], wt * cc[v] * cc[v]);
        }
      }
    }
    __syncthreads();

    // phase B: acc += tril(S) @ Kc
    for (int nt = 0; nt < 4; ++nt) {
      const int nb = cg * 64 + nt * 16;
      v8f c = acc[nt];
      for (int kk = 0; kk < 32; kk += 4)
        c = WMMAF32(frag_rm(Ssh + 16 * rh * SSTR, SSTR, l, hi, kk),
                    frag_cm(Kcur + nb, LDSW, l, hi, kk), c);
      acc[nt] = c;
    }
    buf ^= 1;
  }

  // write raw numerator (normalized + activated in finalize pass)
  float* ob = qraw + ((size_t)b * L_ + t0) * D_;
  for (int nt = 0; nt < 4; ++nt) {
    const int nb = cg * 64 + nt * 16;
    for (int v = 0; v < 8; ++v) {
      const int M = v + 8 * hi;
      ob[(size_t)(16 * rh + M) * D_ + nb + l] = acc[nt][v];
    }
  }
  __syncthreads();
  if (tid < 32)
    dacc[(size_t)b * L_ + t0 + tid] = dsh[tid] + 2.f * esh[tid];
}

// ---------------------------------------------------------------------------
// ||P_prev||_F^2 per batch
// ---------------------------------------------------------------------------
__global__ void qkproj_base(const float* __restrict__ P, float* __restrict__ base) {
  __shared__ float red[256];
  const int b = blockIdx.x;
  const float* Pb = P + (size_t)b * D_ * D_;
  float s = 0.f;
  for (int i = threadIdx.x; i < D_ * D_; i += 256) { float v = Pb[i]; s += v * v; }
  red[threadIdx.x] = s;
  __syncthreads();
  for (int off = 128; off > 0; off >>= 1) {
    if (threadIdx.x < off) red[threadIdx.x] += red[threadIdx.x + off];
    __syncthreads();
  }
  if (threadIdx.x == 0) base[b] = red[0];
}

// sequential cumsum of fro^2 deltas -> 1/(fro + 1e-7)   (4x1024 adds, trivial)
__global__ void qkproj_scan(const float* __restrict__ dacc, const float* __restrict__ base,
                            float* __restrict__ inv) {
  const int b = threadIdx.x;
  if (b >= B_) return;
  float c = base[b];
  for (int t = 0; t < L_; ++t) {
    c += dacc[(size_t)b * L_ + t];
    inv[(size_t)b * L_ + t] = 1.f / (sqrtf(c) + 1e-7f);
  }
}

// q_out = tanh(gain * raw * inv) * output_scale   (in place on d_out)
__global__ void qkproj_finalize(float* __restrict__ qout, const float* __restrict__ inv,
                                const float* __restrict__ lg, const float* __restrict__ osc) {
  const size_t idx = (size_t)blockIdx.x * 256 + threadIdx.x;
  const int i = (int)(idx & (D_ - 1));
  const size_t bl = idx >> 8;
  const float g = expf(lg[i]);
  qout[idx] = tanhf(g * qout[idx] * inv[bl]) * osc[i];
}

// ---------------------------------------------------------------------------
// P_final = P_prev + K^T K per batch (WMMA GEMM, M=N=256, K=1024),
// K chunks double-buffered via async global->LDS.
// ---------------------------------------------------------------------------
__global__ void qkproj_pfinal(const float* __restrict__ k, const float* __restrict__ P,
                              float* __restrict__ Pout)
{
  extern __shared__ float smem2[];
  float* Kbuf0 = smem2;              // 32 x LDSW
  float* Kbuf1 = smem2 + 32 * LDSW;  // 32 x LDSW
  const int b  = blockIdx.y;
  const int i0 = blockIdx.x * 32;
  const int tid = threadIdx.x;
  const int lane = tid & 31, l = lane & 15, hi = lane >> 4;
  const int w = tid >> 5, rh = w >> 2, cg = w & 3;
  const float* Kb = k + (size_t)b * L_ * D_;

  v8f acc[4];
  for (int nt = 0; nt < 4; ++nt) acc[nt] = vzero8();

  float* Kbuf[2] = { Kbuf0, Kbuf1 };
  int buf = 0;
  copy_tile(Kb, Kbuf0, tid);
  for (int tc = 0; tc < L_; tc += 32) {
    wait_async_all();
    __syncthreads();
    if (tc + 32 < L_)
      copy_tile(Kb + (size_t)(tc + 32) * D_, Kbuf[buf ^ 1], tid);
    const float* Kcur = Kbuf[buf];
    for (int nt = 0; nt < 4; ++nt) {
      const int nb = cg * 64 + nt * 16;
      v8f c = acc[nt];
      for (int kk = 0; kk < 32; kk += 4)
        c = WMMAF32(frag_cm(Kcur + i0 + 16 * rh, LDSW, l, hi, kk),
                    frag_cm(Kcur + nb, LDSW, l, hi, kk), c);
      acc[nt] = c;
    }
    buf ^= 1;
  }

  const float* Pb = P + (size_t)b * D_ * D_;
  float* Ob = Pout + (size_t)b * D_ * D_;
  for (int nt = 0; nt < 4; ++nt) {
    const int nb = cg * 64 + nt * 16;
    for (int v = 0; v < 8; ++v) {
      const int M = v + 8 * hi;
      size_t idx = (size_t)(i0 + 16 * rh + M) * D_ + nb + l;
      Ob[idx] = acc[nt][v] + Pb[idx];
    }
  }
}

extern "C" void kernel_launch(void* const* d_in, const int* in_sizes, int n_in,
                              void* d_out, int out_size, void* d_ws, size_t ws_size,
                              hipStream_t stream) {
  const float* q  = (const float*)d_in[0];
  const float* k  = (const float*)d_in[1];
  const float* P  = (const float*)d_in[2];
  const float* lg = (const float*)d_in[3];
  const float* os = (const float*)d_in[4];

  float* qout = (float*)d_out;                       // B*L*D floats
  float* Pout = qout + (size_t)B_ * L_ * D_;         // B*D*D floats

  float* ws   = (float*)d_ws;
  float* dacc = ws;                                  // B*L
  float* inv  = ws + (size_t)B_ * L_;                // B*L
  float* base = ws + (size_t)2 * B_ * L_;            // B

  const size_t smem1 = (size_t)(4 * 32 * LDSW + 32 * SSTR + 64) * sizeof(float);
  const size_t smem2 = (size_t)(2 * 32 * LDSW) * sizeof(float);

  qkproj_base<<<B_, 256, 0, stream>>>(P, base);
  qkproj_pass1<<<dim3(L_ / 32, B_), 256, smem1, stream>>>(q, k, P, qout, dacc);
  qkproj_scan<<<1, 32, 0, stream>>>(dacc, base, inv);
  qkproj_finalize<<<(B_ * L_ * D_) / 256, 256, 0, stream>>>(qout, inv, lg, os);
  qkproj_pfinal<<<dim3(D_ / 32, B_), 256, smem2, stream>>>(k, P, Pout);
}